// Part_Structure_69621419868987
// MI455X (gfx1250) — compile-verified
//
#include <hip/hip_runtime.h>
#include <hip/hip_bf16.h>

// Problem constants
#define BATCH 16
#define NCH   12
#define HW    784
#define T     785
#define D     768
#define D1    512
#define LEAK  0.2f

// GEMM tiling
#define MT  112          // 784 = 7 * 112
#define NT  128          // 512 = 4 * 128
#define KT  16
#define ALD 20           // A LDS row stride (16 data + 4 pad) -> conflict-free frag reads
#define BLD 132          // B LDS row stride (128 data + 4 pad)

typedef float v8f __attribute__((ext_vector_type(8)));
typedef float v2f __attribute__((ext_vector_type(2)));

// ---------------------------------------------------------------------------
// Kernel 1: copy hidden_states -> out (float4 vectorized)
// ---------------------------------------------------------------------------
__global__ void copy_hs_kernel(const float4* __restrict__ src,
                               float4* __restrict__ dst, int n4) {
    int i = blockIdx.x * blockDim.x + threadIdx.x;
    if (i < n4) dst[i] = src[i];
}

// ---------------------------------------------------------------------------
// Kernel 2: adj[b,m,k] = mean over 12 heads of weight[b,h,m+1,k+1]
// Streams the 473 MB weight tensor once, coalesced along k (the HBM floor).
// ---------------------------------------------------------------------------
__global__ void adj_kernel(const float* __restrict__ wt, float* __restrict__ adj) {
    size_t idx = (size_t)blockIdx.x * blockDim.x + threadIdx.x;
    const size_t total = (size_t)BATCH * HW * HW;
    if (idx >= total) return;
    int k = (int)(idx % HW);
    int m = (int)((idx / HW) % HW);
    int b = (int)(idx / ((size_t)HW * HW));
    float s = 0.f;
    const size_t headStride = (size_t)T * T;
    const float* base = wt + ((size_t)b * NCH) * headStride
                           + (size_t)(m + 1) * T + (k + 1);
    #pragma unroll
    for (int h = 0; h < NCH; ++h) s += base[h * headStride];
    adj[idx] = s * (1.0f / NCH);
}

// ---------------------------------------------------------------------------
// Kernel 3: M[b,n,j] = sum_c attention_map[b,c,n] * W1[c,j]   (K=12, tiny)
// ---------------------------------------------------------------------------
__global__ void xw1_kernel(const float* __restrict__ am,
                           const float* __restrict__ W1,
                           float* __restrict__ M) {
    size_t idx = (size_t)blockIdx.x * blockDim.x + threadIdx.x;
    const size_t total = (size_t)BATCH * HW * D1;
    if (idx >= total) return;
    int j = (int)(idx % D1);
    int n = (int)((idx / D1) % HW);
    int b = (int)(idx / ((size_t)D1 * HW));
    float s = 0.f;
    const float* amb = am + (size_t)b * NCH * HW + n;
    #pragma unroll
    for (int c = 0; c < NCH; ++c) s += amb[(size_t)c * HW] * W1[c * D1 + j];
    M[idx] = s;
}

// ---------------------------------------------------------------------------
// Kernel 4: basic_index[b] = argmax_n sum_c attention_map[b,c,n]
// ---------------------------------------------------------------------------
__global__ void argmax_kernel(const float* __restrict__ am, int* __restrict__ bi) {
    __shared__ float sval[256];
    __shared__ int   sidx[256];
    int b = blockIdx.x, tid = threadIdx.x;
    const float* amb = am + (size_t)b * NCH * HW;
    float best = -3.402823e38f; int bidx = 0;
    for (int n = tid; n < HW; n += 256) {
        float s = 0.f;
        #pragma unroll
        for (int c = 0; c < NCH; ++c) s += amb[(size_t)c * HW + n];
        if (s > best) { best = s; bidx = n; }
    }
    sval[tid] = best; sidx[tid] = bidx;
    __syncthreads();
    for (int st = 128; st > 0; st >>= 1) {
        if (tid < st) {
            float ov = sval[tid + st]; int oi = sidx[tid + st];
            if (ov > sval[tid] || (ov == sval[tid] && oi < sidx[tid])) {
                sval[tid] = ov; sidx[tid] = oi;
            }
        }
        __syncthreads();
    }
    if (tid == 0) bi[b] = sidx[0];
}

// ---------------------------------------------------------------------------
// Kernel 5: H1 = relu(adj @ M) — LDS-tiled fp32 WMMA GEMM.
//   Block: 256 threads = 8 waves, tile 112x128, K-step 16.
//   Staging: GLOBAL_LOAD_ASYNC_TO_LDS_B128 (ASYNCcnt) — 4 instr/wave/K-tile.
//   Wave w computes N column [w*16, w*16+16) for all 7 M-subtiles:
//   per k4-step: 1 B frag + 7 A frags (ds_load_b64) + 7 x V_WMMA_F32_16X16X4_F32.
// ---------------------------------------------------------------------------
__global__ __launch_bounds__(256) void gemm_relu_wmma(
        const float* __restrict__ adj,
        const float* __restrict__ Mm,
        float* __restrict__ H1) {
    __shared__ float As[MT * ALD];   // [row][k], padded rows
    __shared__ float Bs[KT * BLD];   // [k][n], padded rows

    const int m0 = blockIdx.x * MT;  // 0..6
    const int n0 = blockIdx.y * NT;  // 0..3
    const int b  = blockIdx.z;
    const int tid  = threadIdx.x;
    const int lane = tid & 31;
    const int w    = tid >> 5;       // wave 0..7

    const float* A  = adj + (size_t)b * HW * HW;   // [784,784]
    const float* Bm = Mm  + (size_t)b * HW * D1;   // [784,512]

    // LDS byte offsets (flat shared pointer truncated to LDS aperture offset)
    const unsigned aBase = (unsigned)(size_t)(void*)As;
    const unsigned bBase = (unsigned)(size_t)(void*)Bs;

    // 16B staging chunks: A has 112*4 = 448 chunks, B has 16*16*... = 512 chunks
    const int c0 = (w * 2 + 0) * 32 + lane;   // 0..479
    const int c1 = (w * 2 + 1) * 32 + lane;   // 32..511

    // A chunk c -> row r = c>>2, k-quad kq = (c&3)*4
    const int ar0 = c0 >> 2, ak0 = (c0 & 3) * 4;
    const int ar1 = c1 >> 2, ak1 = (c1 & 3) * 4;
    // B chunk c -> k-row kr = c>>5, col quad n = (c&31)*4
    const int bk0 = c0 >> 5, bn0 = (c0 & 31) * 4;
    const int bk1 = c1 >> 5, bn1 = (c1 & 31) * 4;

    const unsigned aLo0 = aBase + (unsigned)(ar0 * ALD + ak0) * 4u;
    const unsigned aLo1 = aBase + (unsigned)(ar1 * ALD + ak1) * 4u;
    const unsigned bLo0 = bBase + (unsigned)(bk0 * BLD + bn0) * 4u;
    const unsigned bLo1 = bBase + (unsigned)(bk1 * BLD + bn1) * 4u;

    v8f acc[7];
    #pragma unroll
    for (int i = 0; i < 7; ++i) acc[i] = (v8f){0.f,0.f,0.f,0.f,0.f,0.f,0.f,0.f};

    const int l15  = lane & 15;
    const int koff = (lane >> 4) * 2;      // K offset per A/B fragment layout
    const int nCol = w * 16 + l15;         // wave's column inside the 128-wide tile

    for (int k0 = 0; k0 < HW; k0 += KT) {
        // ---- async stage A tile (112x16) : chunks < 448 ----
        if (c0 < 448) {
            unsigned long long ga =
                (unsigned long long)(size_t)(A + (size_t)(m0 + ar0) * HW + (k0 + ak0));
            asm volatile("global_load_async_to_lds_b128 %0, %1, off"
                         :: "v"(aLo0), "v"(ga) : "memory");
        }
        if (c1 < 448) {
            unsigned long long ga =
                (unsigned long long)(size_t)(A + (size_t)(m0 + ar1) * HW + (k0 + ak1));
            asm volatile("global_load_async_to_lds_b128 %0, %1, off"
                         :: "v"(aLo1), "v"(ga) : "memory");
        }
        // ---- async stage B tile (16x128) : exactly 512 chunks ----
        {
            unsigned long long ga =
                (unsigned long long)(size_t)(Bm + (size_t)(k0 + bk0) * D1 + (n0 + bn0));
            asm volatile("global_load_async_to_lds_b128 %0, %1, off"
                         :: "v"(bLo0), "v"(ga) : "memory");
        }
        {
            unsigned long long ga =
                (unsigned long long)(size_t)(Bm + (size_t)(k0 + bk1) * D1 + (n0 + bn1));
            asm volatile("global_load_async_to_lds_b128 %0, %1, off"
                         :: "v"(bLo1), "v"(ga) : "memory");
        }
        asm volatile("s_wait_asynccnt 0x0" ::: "memory");
        __syncthreads();

        // ---- compute on the K16 tile ----
        #pragma unroll
        for (int ks = 0; ks < KT; ks += 4) {
            const int k = ks + koff;
            v2f bf;
            bf.x = Bs[k * BLD + nCol];
            bf.y = Bs[(k + 1) * BLD + nCol];
            #pragma unroll
            for (int i = 0; i < 7; ++i) {
                const int row = i * 16 + l15;
                v2f af = *(const v2f*)&As[row * ALD + k];   // 8B aligned (k even)
                acc[i] = __builtin_amdgcn_wmma_f32_16x16x4_f32(
                    false, af, false, bf, (short)0, acc[i], false, false);
            }
        }
        __syncthreads();   // before next stage overwrites LDS
    }

    // ---- store with relu; C/D layout: VGPR r -> row r (lanes 0-15) / r+8 ----
    float* C = H1 + (size_t)b * HW * D1;
    const int col  = n0 + nCol;
    const int radd = (lane >> 4) * 8;
    #pragma unroll
    for (int i = 0; i < 7; ++i) {
        const int rowBase = m0 + i * 16 + radd;
        #pragma unroll
        for (int r = 0; r < 8; ++r) {
            C[(size_t)(rowBase + r) * D1 + col] = fmaxf(acc[i][r], 0.f);
        }
    }
}

// ---------------------------------------------------------------------------
// Kernel 6: per batch:  r = adj[b, bi, :];  v = r @ H1[b]  (512);
//           sel = leaky(v @ W2) (768);  out[b,0,:] += sel
// ---------------------------------------------------------------------------
__global__ void final_kernel(const float* __restrict__ adj,
                             const float* __restrict__ H1,
                             const float* __restrict__ W2,
                             const int* __restrict__ bi,
                             float* __restrict__ out) {
    __shared__ float rsh[HW];
    __shared__ float vsh[D1];
    const int b = blockIdx.x, tid = threadIdx.x;
    const int sel_row = bi[b];

    const float* arow = adj + (size_t)b * HW * HW + (size_t)sel_row * HW;
    for (int n = tid; n < HW; n += 256) rsh[n] = arow[n];
    __syncthreads();

    const float* Hb = H1 + (size_t)b * HW * D1;
    for (int j = tid; j < D1; j += 256) {
        float s = 0.f;
        for (int n = 0; n < HW; ++n) s = fmaf(rsh[n], Hb[(size_t)n * D1 + j], s);
        vsh[j] = s;
    }
    __syncthreads();

    float* orow = out + (size_t)b * T * D;  // CLS token row
    for (int e = tid; e < D; e += 256) {
        float s = 0.f;
        for (int j = 0; j < D1; ++j) s = fmaf(vsh[j], W2[(size_t)j * D + e], s);
        s = (s > 0.f) ? s : LEAK * s;
        orow[e] += s;
    }
}

// ---------------------------------------------------------------------------
extern "C" void kernel_launch(void* const* d_in, const int* in_sizes, int n_in,
                              void* d_out, int out_size, void* d_ws, size_t ws_size,
                              hipStream_t stream) {
    const float* hs = (const float*)d_in[0];   // [16,785,768]
    const float* am = (const float*)d_in[1];   // [16,12,28,28]
    const float* wt = (const float*)d_in[2];   // [16,12,785,785]
    const float* W1 = (const float*)d_in[3];   // [12,512]
    const float* W2 = (const float*)d_in[4];   // [512,768]
    float* out = (float*)d_out;                // [16,785,768]

    // Workspace layout (~91 MB)
    float* adj = (float*)d_ws;                          // 16*784*784
    float* Mm  = adj + (size_t)BATCH * HW * HW;         // 16*784*512
    float* H1  = Mm  + (size_t)BATCH * HW * D1;         // 16*784*512
    int*   bi  = (int*)(H1 + (size_t)BATCH * HW * D1);  // 16

    // 1. copy hidden_states -> out
    {
        int n4 = (BATCH * T * D) / 4;
        copy_hs_kernel<<<(n4 + 255) / 256, 256, 0, stream>>>(
            (const float4*)hs, (float4*)out, n4);
    }
    // 2. adjacency reduction (HBM floor: 473 MB read)
    {
        size_t total = (size_t)BATCH * HW * HW;
        adj_kernel<<<(unsigned)((total + 255) / 256), 256, 0, stream>>>(wt, adj);
    }
    // 3. M = x @ W1
    {
        size_t total = (size_t)BATCH * HW * D1;
        xw1_kernel<<<(unsigned)((total + 255) / 256), 256, 0, stream>>>(am, W1, Mm);
    }
    // 4. basic_index
    argmax_kernel<<<BATCH, 256, 0, stream>>>(am, bi);
    // 5. H1 = relu(adj @ M) — LDS-tiled WMMA with async global->LDS staging
    {
        dim3 grid(HW / MT, D1 / NT, BATCH);
        gemm_relu_wmma<<<grid, 256, 0, stream>>>(adj, Mm, H1);
    }
    // 6. fused tail: v = adj[bi] @ H1; sel = leaky(v @ W2); out[:,0,:] += sel
    final_kernel<<<BATCH, 256, 0, stream>>>(adj, H1, W2, bi, out);
}